// GNN_Disentangle_35691178230485
// MI455X (gfx1250) — compile-verified
//
#include <hip/hip_runtime.h>
#include <hip/hip_bf16.h>

// ---------------------------------------------------------------------------
// GEARS-like GNN forward for gfx1250 (MI455X).
// Dense GEMMs use V_WMMA_F32_16X16X4_F32 (native fp32 WMMA, exact).
// Edge aggregation uses fp32 global atomics (L2-resident feature buffers).
// ---------------------------------------------------------------------------

#define NG 5000
#define NB 32
#define NN (NG * NB)        // 160000 nodes
#define HH 64
#define EE 3200000

typedef float v2f __attribute__((ext_vector_type(2)));
typedef float v8f __attribute__((ext_vector_type(8)));

// ---------------- WMMA helper: [16 x 64] (LDS) @ [64 x 64] (LDS) -> acc ----
// A layout in LDS: row-major [16][64].  B layout in LDS: row-major [64][64]
// (K x N).  acc[nb] holds the 16x16 f32 D tile for columns nb*16..nb*16+15.
//
// f32 WMMA operand mapping (wave32):
//   A (16x4):  lanes 0-15: M=lane, VGPR0=K0,VGPR1=K1 ; lanes 16-31: K2,K3
//   B (4x16):  VGPR0: K0 row (lanes 0-15) / K2 row (lanes 16-31); VGPR1: K1/K3
//   C/D:       VGPR r: lanes 0-15 -> M=r, lanes 16-31 -> M=r+8; N = lane&15
__device__ __forceinline__ void wmma_16x64x64(const float* __restrict__ A,
                                              const float* __restrict__ B,
                                              v8f acc[4], int lane) {
  const int m  = lane & 15;
  const int kh = (lane >> 4) << 1;  // 0 for lanes 0-15, 2 for lanes 16-31
#pragma unroll
  for (int kb = 0; kb < 16; ++kb) {
    const int k0 = kb * 4 + kh;
    v2f a;
    a.x = A[m * 64 + k0];
    a.y = A[m * 64 + k0 + 1];
#pragma unroll
    for (int nb = 0; nb < 4; ++nb) {
      v2f b;
      b.x = B[k0 * 64 + nb * 16 + m];
      b.y = B[(k0 + 1) * 64 + nb * 16 + m];
      acc[nb] = __builtin_amdgcn_wmma_f32_16x16x4_f32(
          false, a, false, b, (short)0, acc[nb], false, false);
    }
  }
}

__device__ __forceinline__ void acc_zero(v8f acc[4]) {
#pragma unroll
  for (int nb = 0; nb < 4; ++nb) {
#pragma unroll
    for (int r = 0; r < 8; ++r) acc[nb][r] = 0.0f;
  }
}

// Stage 16 rows of a [N x 64] global matrix into LDS [16][64].
__device__ __forceinline__ void stage_rows(const float* __restrict__ src,
                                           float* __restrict__ A, int tb,
                                           int lane) {
#pragma unroll
  for (int j = 0; j < 16; ++j) {
    const float* r = src + (long)(tb + j) * 64 + lane * 2;
    A[j * 64 + lane * 2]     = r[0];
    A[j * 64 + lane * 2 + 1] = r[1];
  }
}

// ---------------- setup: fold rank-1 linears through the concat-linears ----
// c_base[k] = Wg . Wet_top[:,k]        b1[k] = bg . Wet_top[:,k] + bet[k]
// c_pert[k] = Wp . Wpb_bot[:,k]        b2[k] = bp . Wpb_bot[:,k] + bpb[k]
__global__ void fold_kernel(const float* __restrict__ Wp,
                            const float* __restrict__ bp,
                            const float* __restrict__ Wg,
                            const float* __restrict__ bg,
                            const float* __restrict__ Wet,
                            const float* __restrict__ bet,
                            const float* __restrict__ Wpb,
                            const float* __restrict__ bpb,
                            float* __restrict__ C) {
  int k = threadIdx.x;  // 64 threads
  float cb = 0.f, b1 = 0.f, cp = 0.f, b2 = 0.f;
  for (int j = 0; j < 64; ++j) {
    float wt = Wet[j * 64 + k];          // top half of Wet (rows 0..63)
    cb += Wg[j] * wt;
    b1 += bg[j] * wt;
    float wb = Wpb[(64 + j) * 64 + k];   // bottom half of Wpb (rows 64..127)
    cp += Wp[j] * wb;
    b2 += bp[j] * wb;
  }
  C[k]       = cb;
  C[64 + k]  = b1 + bet[k];
  C[128 + k] = cp;
  C[192 + k] = b2 + bpb[k];
}

// ---------------- degree kernels -------------------------------------------
__global__ void deg_init_kernel(float* __restrict__ deg) {
  int i = blockIdx.x * blockDim.x + threadIdx.x;
  if (i < NN) deg[i] = 1.0f;  // self loop
}

__global__ void deg_count_kernel(const long long* __restrict__ ei,
                                 float* __restrict__ deg) {
  int e = blockIdx.x * blockDim.x + threadIdx.x;
  if (e < EE) {
    int dst = (int)ei[EE + e];
    atomicAdd(&deg[dst], 1.0f);
  }
}

// deg -> invdeg (in place), dinv = 1/sqrt(deg)
__global__ void deg_finish_kernel(float* __restrict__ deg,
                                  float* __restrict__ dinv) {
  int i = blockIdx.x * blockDim.x + threadIdx.x;
  if (i < NN) {
    float d = deg[i];
    dinv[i] = 1.0f / sqrtf(d);
    deg[i]  = 1.0f / d;
  }
}

// ---------------- encode: emb renorm + emb_trans + pert_base_trans ---------
// h0 = (concat([x0*Wg+bg, e]) @ Wet + bet) @ Wpb_top + x1*c_pert + b2
//    = (e @ Wet_bot + x0*c_base + b1) @ Wpb_top + x1*c_pert + b2
__global__ void __launch_bounds__(128)
encode_kernel(const float* __restrict__ x, const float* __restrict__ Emb,
              const float* __restrict__ Wet, const float* __restrict__ Wpb,
              const float* __restrict__ C, float* __restrict__ h_out) {
  __shared__ float sW1[64 * 64];    // Wet bottom half (rows 64..127), KxN
  __shared__ float sW2[64 * 64];    // Wpb top half (rows 0..63), KxN
  __shared__ float sC[256];
  __shared__ float sA[4][16 * 64];  // per-wave staging

  const int tid = threadIdx.x;
  for (int i = tid; i < 64 * 64; i += 128) {
    sW1[i] = Wet[(64 + (i >> 6)) * 64 + (i & 63)];
    sW2[i] = Wpb[i];
  }
  for (int i = tid; i < 256; i += 128) sC[i] = C[i];
  __syncthreads();

  const int wave = tid >> 5, lane = tid & 31;
  const int tb = (blockIdx.x * 4 + wave) * 16;  // first node of this tile
  float* A = sA[wave];

  // stage renormalized embedding rows: e = Emb[node % NG] * min(1, 1/max(|e|,eps))
#pragma unroll 1
  for (int j = 0; j < 16; ++j) {
    int gid = (tb + j) % NG;
    const float* row = Emb + (long)gid * 64 + lane * 2;
    float v0 = row[0], v1 = row[1];
    float ss = v0 * v0 + v1 * v1;
#pragma unroll
    for (int msk = 16; msk >= 1; msk >>= 1) ss += __shfl_xor(ss, msk, 32);
    float s = fminf(1.0f, 1.0f / fmaxf(sqrtf(ss), 1e-12f));
    A[j * 64 + lane * 2]     = v0 * s;
    A[j * 64 + lane * 2 + 1] = v1 * s;
  }
  __syncthreads();

  v8f acc[4];
  acc_zero(acc);
  wmma_16x64x64(A, sW1, acc, lane);   // e @ Wet_bot
  __syncthreads();

  const int m = lane & 15;
  const int hi = (lane >> 4) * 8;
#pragma unroll
  for (int nb = 0; nb < 4; ++nb) {
    const int k = nb * 16 + m;
    const float cb = sC[k], b1 = sC[64 + k];
#pragma unroll
    for (int r = 0; r < 8; ++r) {
      const int rowi = r + hi;
      float x0 = x[(long)(tb + rowi) * 2];
      A[rowi * 64 + k] = acc[nb][r] + x0 * cb + b1;  // h (mid)
    }
  }
  __syncthreads();

  v8f acc2[4];
  acc_zero(acc2);
  wmma_16x64x64(A, sW2, acc2, lane);  // h @ Wpb_top
#pragma unroll
  for (int nb = 0; nb < 4; ++nb) {
    const int k = nb * 16 + m;
    const float cp = sC[128 + k], b2 = sC[192 + k];
#pragma unroll
    for (int r = 0; r < 8; ++r) {
      const int rowi = r + hi;
      float x1 = x[(long)(tb + rowi) * 2 + 1];
      h_out[(long)(tb + rowi) * 64 + k] = acc2[nb][r] + x1 * cp + b2;
    }
  }
}

// ---------------- GCN dense part: hW = h @ W; agg_init = hW*invdeg + bias ---
__global__ void __launch_bounds__(128)
gcn_gemm_kernel(const float* __restrict__ h, const float* __restrict__ W,
                const float* __restrict__ b, const float* __restrict__ invdeg,
                float* __restrict__ hw_out, float* __restrict__ agg_out) {
  __shared__ float sW[64 * 64];
  __shared__ float sB[64];
  __shared__ float sA[4][16 * 64];

  const int tid = threadIdx.x;
  for (int i = tid; i < 64 * 64; i += 128) sW[i] = W[i];
  if (tid < 64) sB[tid] = b[tid];
  __syncthreads();

  const int wave = tid >> 5, lane = tid & 31;
  const int tb = (blockIdx.x * 4 + wave) * 16;
  float* A = sA[wave];
  stage_rows(h, A, tb, lane);
  __syncthreads();

  v8f acc[4];
  acc_zero(acc);
  wmma_16x64x64(A, sW, acc, lane);

  const int m = lane & 15;
  const int hi = (lane >> 4) * 8;
#pragma unroll
  for (int nb = 0; nb < 4; ++nb) {
    const int k = nb * 16 + m;
    const float bias = sB[k];
#pragma unroll
    for (int r = 0; r < 8; ++r) {
      const int rowi = r + hi;
      const long idx = (long)(tb + rowi) * 64 + k;
      float v = acc[nb][r];
      hw_out[idx] = v;
      agg_out[idx] = v * invdeg[tb + rowi] + bias;  // self-loop term + bias
    }
  }
}

// ---------------- edge scatter: agg[dst] += hW[src] * dinv[src]*dinv[dst] --
// 4 threads per edge, 16 floats each (float4 gathers, fp32 atomics).
__global__ void scatter_kernel(const long long* __restrict__ ei,
                               const float* __restrict__ dinv,
                               const float* __restrict__ hw,
                               float* __restrict__ agg) {
  int idx = blockIdx.x * blockDim.x + threadIdx.x;  // < 4*EE
  int e = idx >> 2;
  int q = (idx & 3) * 16;
  if (e + 2048 < EE) __builtin_prefetch(&ei[e + 2048], 0, 0);
  int src = (int)ei[e];
  int dst = (int)ei[EE + e];
  float c = dinv[src] * dinv[dst];
  const float4* s = (const float4*)(hw + (long)src * 64 + q);
  float* d = agg + (long)dst * 64 + q;
#pragma unroll
  for (int t = 0; t < 4; ++t) {
    float4 v = s[t];
    atomicAdd(d + t * 4 + 0, v.x * c);
    atomicAdd(d + t * 4 + 1, v.y * c);
    atomicAdd(d + t * 4 + 2, v.z * c);
    atomicAdd(d + t * 4 + 3, v.w * c);
  }
}

// ---------------- elementwise: h = relu(agg) or copy ------------------------
__global__ void act_kernel(const float* __restrict__ agg,
                           float* __restrict__ h, int do_relu) {
  long i = (long)blockIdx.x * blockDim.x + threadIdx.x;
  if (i < (long)NN * 64) {
    float v = agg[i];
    h[i] = do_relu ? fmaxf(v, 0.0f) : v;
  }
}

// ---------------- MLP pass 1: part = relu(h@Wr1[:, :64] + br1[:64]) @ Wr2[:64]
__global__ void __launch_bounds__(128)
mlp1_kernel(const float* __restrict__ h, const float* __restrict__ Wr1,
            const float* __restrict__ br1, const float* __restrict__ Wr2,
            float* __restrict__ part) {
  __shared__ float sW1[64 * 64];   // Wr1 columns 0..63
  __shared__ float sW2[64 * 64];   // Wr2 rows 0..63
  __shared__ float sB1[64];
  __shared__ float sA[4][16 * 64];

  const int tid = threadIdx.x;
  for (int i = tid; i < 64 * 64; i += 128) {
    sW1[i] = Wr1[(i >> 6) * 128 + (i & 63)];
    sW2[i] = Wr2[i];
  }
  if (tid < 64) sB1[tid] = br1[tid];
  __syncthreads();

  const int wave = tid >> 5, lane = tid & 31;
  const int tb = (blockIdx.x * 4 + wave) * 16;
  float* A = sA[wave];
  stage_rows(h, A, tb, lane);
  __syncthreads();

  v8f acc[4];
  acc_zero(acc);
  wmma_16x64x64(A, sW1, acc, lane);
  __syncthreads();

  const int m = lane & 15;
  const int hi = (lane >> 4) * 8;
#pragma unroll
  for (int nb = 0; nb < 4; ++nb) {
    const int k = nb * 16 + m;
    const float bias = sB1[k];
#pragma unroll
    for (int r = 0; r < 8; ++r) {
      const int rowi = r + hi;
      A[rowi * 64 + k] = fmaxf(acc[nb][r] + bias, 0.0f);  // r1 low half
    }
  }
  __syncthreads();

  v8f acc2[4];
  acc_zero(acc2);
  wmma_16x64x64(A, sW2, acc2, lane);
#pragma unroll
  for (int nb = 0; nb < 4; ++nb) {
    const int k = nb * 16 + m;
#pragma unroll
    for (int r = 0; r < 8; ++r) {
      const int rowi = r + hi;
      part[(long)(tb + rowi) * 64 + k] = acc2[nb][r];
    }
  }
}

// ---------------- MLP pass 2: add high half, relu, 64->1 dot, basal add ----
__global__ void __launch_bounds__(128)
mlp2_kernel(const float* __restrict__ h, const float* __restrict__ part,
            const float* __restrict__ Wr1, const float* __restrict__ br1,
            const float* __restrict__ Wr2, const float* __restrict__ br2,
            const float* __restrict__ Wr3, const float* __restrict__ br3,
            const float* __restrict__ x, float* __restrict__ out) {
  __shared__ float sW1[64 * 64];   // Wr1 columns 64..127
  __shared__ float sW2[64 * 64];   // Wr2 rows 64..127
  __shared__ float sB1[64];
  __shared__ float sB2[64];
  __shared__ float sW3[64];
  __shared__ float sA[4][16 * 64];

  const int tid = threadIdx.x;
  for (int i = tid; i < 64 * 64; i += 128) {
    sW1[i] = Wr1[(i >> 6) * 128 + 64 + (i & 63)];
    sW2[i] = Wr2[(64 + (i >> 6)) * 64 + (i & 63)];
  }
  if (tid < 64) {
    sB1[tid] = br1[64 + tid];
    sB2[tid] = br2[tid];
    sW3[tid] = Wr3[tid];
  }
  __syncthreads();

  const int wave = tid >> 5, lane = tid & 31;
  const int tb = (blockIdx.x * 4 + wave) * 16;
  float* A = sA[wave];
  stage_rows(h, A, tb, lane);
  __syncthreads();

  v8f acc[4];
  acc_zero(acc);
  wmma_16x64x64(A, sW1, acc, lane);
  __syncthreads();

  const int m = lane & 15;
  const int hi = (lane >> 4) * 8;
#pragma unroll
  for (int nb = 0; nb < 4; ++nb) {
    const int k = nb * 16 + m;
    const float bias = sB1[k];
#pragma unroll
    for (int r = 0; r < 8; ++r) {
      const int rowi = r + hi;
      A[rowi * 64 + k] = fmaxf(acc[nb][r] + bias, 0.0f);  // r1 high half
    }
  }
  __syncthreads();

  v8f acc2[4];
  acc_zero(acc2);
  wmma_16x64x64(A, sW2, acc2, lane);
  __syncthreads();

  // r2 = relu(part + acc2 + br2); stash tile in LDS for per-row dot
#pragma unroll
  for (int nb = 0; nb < 4; ++nb) {
    const int k = nb * 16 + m;
    const float bias = sB2[k];
#pragma unroll
    for (int r = 0; r < 8; ++r) {
      const int rowi = r + hi;
      float v = acc2[nb][r] + part[(long)(tb + rowi) * 64 + k] + bias;
      A[rowi * 64 + k] = fmaxf(v, 0.0f);
    }
  }
  __syncthreads();

  if (lane < 16) {
    const int node = tb + lane;
    float dot = 0.0f;
#pragma unroll
    for (int k = 0; k < 64; ++k) dot += A[lane * 64 + k] * sW3[k];
    out[node] = dot + br3[0] + x[(long)node * 2];
  }
}

// ---------------------------------------------------------------------------
extern "C" void kernel_launch(void* const* d_in, const int* in_sizes, int n_in,
                              void* d_out, int out_size, void* d_ws,
                              size_t ws_size, hipStream_t stream) {
  (void)in_sizes; (void)n_in; (void)out_size; (void)ws_size;

  const float*     x    = (const float*)d_in[0];
  const long long* ei   = (const long long*)d_in[1];
  const float*     Wp   = (const float*)d_in[2];
  const float*     bp   = (const float*)d_in[3];
  const float*     Wg   = (const float*)d_in[4];
  const float*     bg   = (const float*)d_in[5];
  const float*     Emb  = (const float*)d_in[6];
  const float*     Wet  = (const float*)d_in[7];
  const float*     bet  = (const float*)d_in[8];
  const float*     Wpb  = (const float*)d_in[9];
  const float*     bpb  = (const float*)d_in[10];
  const float*     Wc1  = (const float*)d_in[11];
  const float*     bc1  = (const float*)d_in[12];
  const float*     Wc2  = (const float*)d_in[13];
  const float*     bc2  = (const float*)d_in[14];
  const float*     Wr1  = (const float*)d_in[15];
  const float*     br1  = (const float*)d_in[16];
  const float*     Wr2  = (const float*)d_in[17];
  const float*     br2  = (const float*)d_in[18];
  const float*     Wr3  = (const float*)d_in[19];
  const float*     br3  = (const float*)d_in[20];
  float* out = (float*)d_out;

  // workspace layout (all 256B aligned)
  char* ws = (char*)d_ws;
  float* consts  = (float*)(ws);                               // 4*64 floats
  float* deg     = (float*)(ws + 1024);                        // NN floats (becomes invdeg)
  float* dinv    = (float*)(ws + 1024 + (size_t)NN * 4);       // NN floats
  size_t fo = 1024 + (size_t)NN * 8;
  float* buf_h   = (float*)(ws + fo);                          // NN*64
  float* buf_t   = (float*)(ws + fo + (size_t)NN * 64 * 4);    // NN*64
  float* buf_agg = (float*)(ws + fo + (size_t)NN * 64 * 8);    // NN*64

  const int TILES = NN / 16;        // 10000
  const int GBLK  = TILES / 4;      // 2500 blocks, 4 waves each

  fold_kernel<<<1, 64, 0, stream>>>(Wp, bp, Wg, bg, Wet, bet, Wpb, bpb, consts);

  deg_init_kernel<<<(NN + 255) / 256, 256, 0, stream>>>(deg);
  deg_count_kernel<<<(EE + 255) / 256, 256, 0, stream>>>(ei, deg);
  deg_finish_kernel<<<(NN + 255) / 256, 256, 0, stream>>>(deg, dinv);

  encode_kernel<<<GBLK, 128, 0, stream>>>(x, Emb, Wet, Wpb, consts, buf_h);

  // GCN layer 1 (+relu)
  gcn_gemm_kernel<<<GBLK, 128, 0, stream>>>(buf_h, Wc1, bc1, deg, buf_t, buf_agg);
  scatter_kernel<<<(EE * 4) / 256, 256, 0, stream>>>(ei, dinv, buf_t, buf_agg);
  act_kernel<<<(NN * 64) / 256, 256, 0, stream>>>(buf_agg, buf_h, 1);

  // GCN layer 2 (no relu)
  gcn_gemm_kernel<<<GBLK, 128, 0, stream>>>(buf_h, Wc2, bc2, deg, buf_t, buf_agg);
  scatter_kernel<<<(EE * 4) / 256, 256, 0, stream>>>(ei, dinv, buf_t, buf_agg);
  act_kernel<<<(NN * 64) / 256, 256, 0, stream>>>(buf_agg, buf_h, 0);

  // recovery MLP (split into two halves to keep LDS <= 48KB per kernel)
  mlp1_kernel<<<GBLK, 128, 0, stream>>>(buf_h, Wr1, br1, Wr2, buf_t);
  mlp2_kernel<<<GBLK, 128, 0, stream>>>(buf_h, buf_t, Wr1, br1, Wr2, br2, Wr3,
                                        br3, x, out);
}